// LossForBERT_48954037240087
// MI455X (gfx1250) — compile-verified
//
#include <hip/hip_runtime.h>
#include <hip/hip_bf16.h>

typedef __attribute__((ext_vector_type(2))) float v2f;
typedef __attribute__((ext_vector_type(8))) float v8f;

#define NN 16384
#define HH 128
#define KNEG 64

// ---------------------------------------------------------------------------
// Main kernel: one block handles a tile of 16 consecutive n's.
//   8 waves per block; wave w handles negative slots {w, w+8, ..., w+56}.
//   Each wave also computes the positive slot for its tile (cheap, avoids
//   cross-wave synchronization of pos scores).
// Per slot: chain of 32 x V_WMMA_F32_16X16X4_F32 accumulating
//   C[m][n] = sum_h P[m][h] * E[n][h]; we consume only the diagonal.
// ---------------------------------------------------------------------------
__global__ __launch_bounds__(256) void nce_loss_main(
    const float* __restrict__ pred,      // [N,H]
    const float* __restrict__ ww,        // [V,H]
    const float* __restrict__ bias,      // [V]
    const float* __restrict__ lw,        // [N]
    const int*   __restrict__ lids,      // [N]
    const int*   __restrict__ nids,      // [K,N]
    float*       __restrict__ partials)  // [gridDim.x]
{
    __shared__ float lds_pred[16 * HH];  // 8 KB pred tile, shared by all waves
    __shared__ float lds_wsum[8];

    const int tid  = (int)threadIdx.x;
    const int lane = tid & 31;
    const int wave = tid >> 5;
    const int n0   = (int)blockIdx.x * 16;

    // ---- cooperative, coalesced stage of the 16x128 pred tile into LDS ----
    {
        const float4* src = (const float4*)(pred + (size_t)n0 * HH);
        float4*       dst = (float4*)lds_pred;
        #pragma unroll
        for (int i = tid; i < 16 * HH / 4; i += 256) dst[i] = src[i];
    }
    __syncthreads();

    // Fragment addressing for V_WMMA_F32_16X16X4_F32 (wave32):
    //   A (16x4):  lane L holds row M=L&15, K-cols {2*(L>>4), 2*(L>>4)+1}
    //   B (4x16):  lane L holds col N=L&15, K-rows {2*(L>>4), 2*(L>>4)+1}
    const int row  = lane & 15;          // A row / B column owned by this lane
    const int hi2  = (lane >> 4) * 2;    // K sub-offset (0 or 2)

    // Preload all 32 A fragments (whole 16x128 tile striped across the wave):
    // 64 VGPRs, reused across all 9 slot chains this wave runs.
    v2f a[32];
    #pragma unroll
    for (int j = 0; j < 32; ++j) {
        const float* p = &lds_pred[row * HH + 4 * j + hi2];
        v2f t; t.x = p[0]; t.y = p[1];
        a[j] = t;
    }

    const int colN = n0 + (lane & 15);               // column index for B gather
    const bool act = (lane < 8) || (lane >= 24);     // lanes that own a diag entry
    const int  myn = n0 + ((lane < 8) ? lane : (lane - 16)); // valid when act

    // score chain: C += A_j x B_j over H, B rows gathered from one vocab row
    auto dotchain = [&](const float* __restrict__ erow) -> v8f {
        v8f c = {};
        #pragma unroll
        for (int j = 0; j < 32; ++j) {
            v2f b = *(const v2f*)(erow + 4 * j + hi2);   // global_load_b64 gather
            c = __builtin_amdgcn_wmma_f32_16x16x4_f32(
                    false, a[j], false, b, (short)0, c, false, false);
        }
        return c;
    };

    // diagonal extraction from C/D layout:
    //   lanes 0..7  -> C[lane]       (M = lane)
    //   lanes 24..31-> C[lane - 24]  (M = lane - 16)
    auto diag = [&](v8f c) -> float {
        float carr[8];
        #pragma unroll
        for (int i = 0; i < 8; ++i) carr[i] = c[i];
        const int idx = (lane < 8) ? lane : (lane - 24);
        float d = 0.f;
        #pragma unroll
        for (int i = 0; i < 8; ++i) d = (idx == i) ? carr[i] : d;
        return d;
    };

    // ---- positive slot ----
    const int  pcol_id = lids[colN];
    const v8f  cpos    = dotchain(ww + (size_t)pcol_id * HH);
    float posv = 0.f, w = 0.f;
    if (act) {
        const int myid = lids[myn];
        posv = diag(cpos) + bias[myid];
        w    = lw[myn];
    }

    // ---- this wave's 8 negative slots ----
    float acc = 0.f;
    for (int k = wave; k < KNEG; k += 8) {           // wave-uniform loop
        const int ncol_id = nids[k * NN + colN];
        const v8f cn      = dotchain(ww + (size_t)ncol_id * HH);
        if (act) {
            const int   mynid = nids[k * NN + myn];
            const float negv  = diag(cn) + bias[mynid];
            const float x     = negv - posv;          // softplus(neg - pos)
            const float sp    = fmaxf(x, 0.f) + log1pf(__expf(-fabsf(x)));
            acc += sp * w;
        }
    }

    // ---- deterministic reduction: wave shuffle -> LDS -> block partial ----
    #pragma unroll
    for (int off = 16; off > 0; off >>= 1)
        acc += __shfl_xor(acc, off, 32);
    if (lane == 0) lds_wsum[wave] = acc;
    __syncthreads();
    if (tid == 0) {
        float s = 0.f;
        #pragma unroll
        for (int i = 0; i < 8; ++i) s += lds_wsum[i];
        partials[blockIdx.x] = s;
    }
}

// ---------------------------------------------------------------------------
// Finalize: sum block partials + label_weights (fixed order -> deterministic),
// then loss = S / (K * sum(lw)).
// ---------------------------------------------------------------------------
__global__ __launch_bounds__(256) void nce_loss_finalize(
    const float* __restrict__ partials, int nparts,
    const float* __restrict__ lw,
    float*       __restrict__ out)
{
    __shared__ float red[256];
    const int tid = (int)threadIdx.x;

    float s = 0.f;
    for (int i = tid; i < nparts; i += 256) s += partials[i];
    float ws = 0.f;
    for (int i = tid; i < NN; i += 256) ws += lw[i];

    red[tid] = s;
    __syncthreads();
    for (int o = 128; o > 0; o >>= 1) {
        if (tid < o) red[tid] += red[tid + o];
        __syncthreads();
    }
    const float S = red[0];
    __syncthreads();

    red[tid] = ws;
    __syncthreads();
    for (int o = 128; o > 0; o >>= 1) {
        if (tid < o) red[tid] += red[tid + o];
        __syncthreads();
    }
    if (tid == 0) out[0] = S / ((float)KNEG * red[0]);
}

extern "C" void kernel_launch(void* const* d_in, const int* in_sizes, int n_in,
                              void* d_out, int out_size, void* d_ws, size_t ws_size,
                              hipStream_t stream) {
    (void)in_sizes; (void)n_in; (void)out_size; (void)ws_size;

    const float* pred = (const float*)d_in[0];   // [N,H] f32
    const float* ww   = (const float*)d_in[1];   // [V,H] f32
    const float* bias = (const float*)d_in[2];   // [V]   f32
    const float* lw   = (const float*)d_in[3];   // [N]   f32
    const int*   lids = (const int*)d_in[4];     // [N]   i32
    const int*   nids = (const int*)d_in[5];     // [K,N] i32
    float*       out  = (float*)d_out;
    float*       part = (float*)d_ws;            // 1024 floats of scratch

    const int nblocks = NN / 16;                 // 1024 tiles
    nce_loss_main<<<nblocks, 256, 0, stream>>>(pred, ww, bias, lw, lids, nids, part);
    nce_loss_finalize<<<1, 256, 0, stream>>>(part, nblocks, lw, out);
}